// TaskAttention_79370995630691
// MI455X (gfx1250) — compile-verified
//
#include <hip/hip_runtime.h>

typedef __attribute__((ext_vector_type(2))) float v2f;
typedef __attribute__((ext_vector_type(4))) float v4f;
typedef __attribute__((ext_vector_type(8))) float v8f;

#define S_LEN  8192
#define BATCH  64
#define DDIM   128
#define TILE_S 64     // rows of q per workgroup (4 waves x 16 rows)
#define PITCH  132    // LDS row pitch in floats; 132 % 64 != 0 -> no bank conflicts on column reads
#define NDROP  819    // max(int(8192 * 0.1), 1)

// ---------------------------------------------------------------------------
// Phase 1: batched GEMV  w[b,s] = sum_d q[s,b,d] * k[b,d]   via f32 WMMA.
// Grid: (BATCH, S_LEN/TILE_S), block 128 threads (4 waves).
// Each wave computes 16 consecutive s-rows: A = 16x4 q tile, B = k chunk
// broadcast into all 16 columns, C accumulates over D in steps of 4.
// ---------------------------------------------------------------------------
__global__ __launch_bounds__(128) void
tattn_score_kernel(const float* __restrict__ q, const float* __restrict__ kvec,
                   float* __restrict__ w)
{
    __shared__ float lq[TILE_S * PITCH];
    __shared__ float lk[DDIM];

    const int bb   = blockIdx.x;      // batch
    const int st   = blockIdx.y;      // s-tile
    const int tid  = threadIdx.x;
    const int lane = tid & 31;
    const int wv   = tid >> 5;        // wave id 0..3

    // k vector for this batch (128 floats, one per thread -> no divergence)
    lk[tid] = kvec[bb * DDIM + tid];

    // Stage q tile: each wave loads 16 rows; per row 32 lanes x float4 = 512B
    // contiguous (q row-major over d). Non-temporal: q (268MB) exceeds L2.
    for (int rr = 0; rr < 16; ++rr) {
        const int r = wv * 16 + rr;
        const int s = st * TILE_S + r;
        const v4f* src = (const v4f*)(q + ((size_t)s * BATCH + bb) * DDIM);
        v4f val = __builtin_nontemporal_load(src + lane);
        *(v4f*)(&lq[r * PITCH + lane * 4]) = val;
    }
    __syncthreads();

    // WMMA A layout (32-bit, 16x4): lanes 0-15 -> M=lane, VGPR0=K0, VGPR1=K1;
    // lanes 16-31 -> M=lane-16, VGPR0=K2, VGPR1=K3.  Same striping for B rows.
    const int   m0   = wv * 16;
    const int   koff = (lane >> 4) << 1;               // 0 or 2
    const float* arow = &lq[(m0 + (lane & 15)) * PITCH];

    v8f c = {};
    for (int d = 0; d < DDIM; d += 4) {
        v2f a, b;
        a.x = arow[d + koff];
        a.y = arow[d + koff + 1];
        b.x = lk[d + koff];          // broadcast: every B column = k chunk
        b.y = lk[d + koff + 1];
        c = __builtin_amdgcn_wmma_f32_16x16x4_f32(
                /*neg_a=*/false, a, /*neg_b=*/false, b,
                /*c_mod=*/(short)0, c, /*reuse_a=*/false, /*reuse_b=*/false);
    }

    // D layout: VGPR v holds (M = v + 8*(lane>=16), N = lane&15); every column
    // identical.  Lane (lane&15)==v writes row v (+8 for the upper half).
    const int half = (lane >> 4) * 8;
    #pragma unroll
    for (int v = 0; v < 8; ++v) {
        if ((lane & 15) == v) {
            const int srow = st * TILE_S + m0 + v + half;
            w[(size_t)bb * S_LEN + srow] = c[v];
        }
    }
}

// ---------------------------------------------------------------------------
// Phase 2: per batch, drop the NDROP indices with largest (gumbel - w)
// (== top-k of log_softmax(mx - w) + g, since per-batch constants don't
// affect ranking).  One workgroup per batch; keys in LDS; exact 32-bit
// bitwise radix-select; deterministic tie resolution; exactly NDROP zeros.
// ---------------------------------------------------------------------------
__global__ __launch_bounds__(256) void
tattn_mask_kernel(const float* __restrict__ w, float* __restrict__ out)
{
    __shared__ unsigned      keys[S_LEN];   // 32 KB
    __shared__ unsigned char drop[S_LEN];   // 8 KB
    __shared__ int cnt;
    __shared__ int eqcnt;

    const int bb  = blockIdx.x;
    const int tid = threadIdx.x;

    // Build sortable keys: key = gumbel(b,s) - w[b,s], mapped monotonically
    // to uint32 (larger float -> larger uint).
    for (int i = tid; i < S_LEN; i += 256) {
        float wvv = w[(size_t)bb * S_LEN + i];
        unsigned h = (unsigned)(i * BATCH + bb) * 0x9E3779B9u + 0x7F4A7C15u;
        h ^= h >> 16; h *= 0x85EBCA6Bu; h ^= h >> 13; h *= 0xC2B2AE35u; h ^= h >> 16;
        float u = ((float)(h >> 8) + 0.5f) * (1.0f / 16777216.0f);   // (0,1)
        float g = -__logf(-__logf(u));                               // Gumbel(0,1)
        float key = g - wvv;
        unsigned ku = __float_as_uint(key);
        ku = (ku & 0x80000000u) ? ~ku : (ku | 0x80000000u);
        keys[i] = ku;
    }
    __syncthreads();

    // Radix-select the NDROP-th largest key.
    unsigned prefix = 0;
    int remaining = NDROP;
    for (int bit = 31; bit >= 0; --bit) {
        if (tid == 0) cnt = 0;
        __syncthreads();
        const unsigned bmask  = 1u << bit;
        const unsigned himask = ~(bmask - 1u);
        const unsigned want   = prefix | bmask;
        int local = 0;
        for (int i = tid; i < S_LEN; i += 256)
            local += ((keys[i] & himask) == want) ? 1 : 0;
        atomicAdd(&cnt, local);
        __syncthreads();
        const int c = cnt;
        if (c >= remaining) prefix |= bmask;
        else                remaining -= c;
        __syncthreads();
    }
    const unsigned thresh = prefix;   // value of NDROP-th largest
    // remaining == number of keys equal to thresh that must also be dropped.

    if (tid == 0) eqcnt = 0;
    __syncthreads();
    int local_eq = 0;
    for (int i = tid; i < S_LEN; i += 256) {
        const unsigned kv = keys[i];
        drop[i] = (kv > thresh) ? 1 : 0;
        local_eq += (kv == thresh) ? 1 : 0;
    }
    atomicAdd(&eqcnt, local_eq);
    __syncthreads();

    const int eq = eqcnt;
    if (eq == remaining) {            // common case: unique threshold value
        for (int i = tid; i < S_LEN; i += 256)
            if (keys[i] == thresh) drop[i] = 1;
    } else if (tid == 0) {            // rare ties: deterministic by index
        int left = remaining;
        for (int i = 0; i < S_LEN && left > 0; ++i)
            if (keys[i] == thresh) { drop[i] = 1; --left; }
    }
    __syncthreads();

    // Output [S, B, 1]: out[s*B + b]
    for (int i = tid; i < S_LEN; i += 256)
        out[(size_t)i * BATCH + bb] = drop[i] ? 0.0f : 1.0f;
}

// ---------------------------------------------------------------------------
extern "C" void kernel_launch(void* const* d_in, const int* in_sizes, int n_in,
                              void* d_out, int out_size, void* d_ws, size_t ws_size,
                              hipStream_t stream)
{
    (void)in_sizes; (void)n_in; (void)out_size; (void)ws_size;
    const float* q    = (const float*)d_in[0];   // [S, B, D] f32
    const float* kvec = (const float*)d_in[1];   // [B, D, 1] f32
    // d_in[2] = lengths (int64) -- unused by the reference computation.

    float* w = (float*)d_ws;                     // [B, S] scores, 2 MB scratch

    dim3 g1(BATCH, S_LEN / TILE_S);
    tattn_score_kernel<<<g1, 128, 0, stream>>>(q, kvec, w);
    tattn_mask_kernel<<<BATCH, 256, 0, stream>>>(w, (float*)d_out);
}